// DVSA_41446434406676
// MI455X (gfx1250) — compile-verified
//
#include <hip/hip_runtime.h>
#include <hip/hip_bf16.h>
#include <math.h>

#define EPS   1e-05f
#define DELTA 0.2f

typedef __attribute__((ext_vector_type(2))) float v2f;
typedef __attribute__((ext_vector_type(8))) float v8f;

// Problem constants (from reference setup_inputs)
#define NA 16
#define NS 32
#define NB 100
#define NE 12
#define DD 512
#define NCOL (NA * NE)        // 192
#define NROW (NA * NS * NB)   // 51200
#define NAS  (NA * NS)        // 512

#define BS_GSTRIDE 193        // padded (192+1): group stride ≡ 4 mod 64 banks
#define AS_RSTRIDE 34         // padded (32+2), even => aligned b64 frag loads

// ---------------------------------------------------------------------------
// Kernel 1: S_ = vis @ word^T with column masking, fp32 WMMA 16x16x4.
// Block = 256 threads (8 waves). Each wave owns a 16-row strip; block covers
// 128 rows; each wave computes all 12 column tiles (N = 192).
// LDS tiles are stored "fragment-ready": each lane's (K, K+1) pair is one
// aligned 8-byte word -> single ds_load_b64 straight into an even VGPR pair
// (no mov shuffling before v_wmma). Next A chunk is prefetched
// (global_prefetch_b8) to hide HBM latency behind the WMMA burst.
// ---------------------------------------------------------------------------
__global__ __launch_bounds__(256)
void k_gemm(const float* __restrict__ V, const float* __restrict__ W,
            const int* __restrict__ ent, float* __restrict__ S)
{
    // B tile: 8 k-groups x 192 cols x 4 K-values (padded group stride)
    __shared__ float Bs[8 * BS_GSTRIDE * 4];
    // A tile: per-wave 16 rows x 32 K (padded row stride, even)
    __shared__ float As[8][16 * AS_RSTRIDE];

    const int tid  = threadIdx.x;
    const int wave = tid >> 5;
    const int lane = tid & 31;
    const int m16  = lane & 15;
    const int half = lane >> 4;        // 0: lanes 0-15, 1: lanes 16-31
    const int kb   = half * 2;         // K sub-offset for A/B fragments
    const int rowBase = blockIdx.x * 128 + wave * 16;

    v8f acc[12] = {};

    for (int k0 = 0; k0 < DD; k0 += 32) {
        // Stage B tile: Bs[g][n][kk&3] = W[n*512 + k0 + kk], coalesced reads.
        for (int i = tid; i < 32 * NCOL; i += 256) {
            int n  = i >> 5;
            int kk = i & 31;
            Bs[((kk >> 2) * BS_GSTRIDE + n) * 4 + (kk & 3)] =
                W[(size_t)n * DD + k0 + kk];
        }
        // Stage A tile for this wave (16 rows x 32 K), coalesced; prefetch
        // the next K-chunk while WMMAs run.
        for (int r = 0; r < 16; ++r) {
            const float* src = &V[(size_t)(rowBase + r) * DD + k0 + lane];
            As[wave][r * AS_RSTRIDE + lane] = *src;
            if (k0 + 32 < DD) __builtin_prefetch(src + 32, 0, 1);
        }
        __syncthreads();

#pragma unroll
        for (int g = 0; g < 8; ++g) {
            v2f a = *(const v2f*)&As[wave][m16 * AS_RSTRIDE + g * 4 + kb];
            const float* bb = &Bs[(g * BS_GSTRIDE) * 4 + kb];
#pragma unroll
            for (int nt = 0; nt < 12; ++nt) {
                v2f b = *(const v2f*)&bb[(nt * 16 + m16) * 4];
                acc[nt] = __builtin_amdgcn_wmma_f32_16x16x4_f32(
                    false, a, false, b, (short)0, acc[nt], false, false);
            }
        }
        __syncthreads();
    }

    // Store with entity-length column masking.
#pragma unroll
    for (int nt = 0; nt < 12; ++nt) {
        int j  = nt * 16 + m16;
        int a2 = j / NE;
        int e  = j - a2 * NE;
        float mask = (e >= ent[a2]) ? 0.0f : 1.0f;
#pragma unroll
        for (int r = 0; r < 8; ++r) {
            int m = r + half * 8;
            S[(size_t)(rowBase + m) * NCOL + j] = acc[nt][r] * mask;
        }
    }
}

// ---------------------------------------------------------------------------
// Kernel 2: max/argmax over Nb -> D_ind (as float) and D_sim, straight to out.
// ---------------------------------------------------------------------------
__global__ void k_rowmax(const float* __restrict__ S, float* __restrict__ out)
{
    int t = blockIdx.x * 256 + threadIdx.x;
    if (t >= NAS * NCOL) return;
    int as = t / NCOL;
    int j  = t - as * NCOL;
    const float* p = S + (size_t)as * NB * NCOL + j;
    float mx = p[0];
    int   mi = 0;
    for (int b = 1; b < NB; ++b) {
        float v = p[(size_t)b * NCOL];
        if (v > mx) { mx = v; mi = b; }   // strict > keeps first occurrence
    }
    out[t]               = (float)mi;     // D_ind
    out[NAS * NCOL + t]  = mx;            // D_sim
}

// ---------------------------------------------------------------------------
// Kernel 3: 1/(||vis_row_i|| + EPS) for i in [0,100)
// ---------------------------------------------------------------------------
__global__ void k_norms(const float* __restrict__ V, float* __restrict__ ninv)
{
    __shared__ float red[256];
    int i = blockIdx.x;
    float s = 0.0f;
    for (int d = threadIdx.x; d < DD; d += 256) {
        float v = V[(size_t)i * DD + d];
        s += v * v;
    }
    red[threadIdx.x] = s;
    __syncthreads();
    for (int o = 128; o > 0; o >>= 1) {
        if (threadIdx.x < o) red[threadIdx.x] += red[threadIdx.x + o];
        __syncthreads();
    }
    if (threadIdx.x == 0) ninv[i] = 1.0f / (sqrtf(red[0]) + EPS);
}

// ---------------------------------------------------------------------------
// Kernel 4: Gram matrix of normalized rows 0..99
// ---------------------------------------------------------------------------
__global__ void k_gram(const float* __restrict__ V, const float* __restrict__ ninv,
                       float* __restrict__ G)
{
    int t = blockIdx.x * 256 + threadIdx.x;
    if (t >= NB * NB) return;
    int i = t / NB, j = t - i * NB;
    const float* pi = V + (size_t)i * DD;
    const float* pj = V + (size_t)j * DD;
    float s = 0.0f;
    for (int d = 0; d < DD; ++d) s += pi[d] * pj[d];
    G[t] = s * ninv[i] * ninv[j];
}

// ---------------------------------------------------------------------------
// Kernel 5: per-(a,e) min-max normalization of diag sim scores + argmax idx
// ---------------------------------------------------------------------------
__global__ void k_simn(const float* __restrict__ out,
                       float* __restrict__ simn, int* __restrict__ bidx)
{
    int t = blockIdx.x * blockDim.x + threadIdx.x;
    if (t >= NA * NE) return;
    int a = t / NE, e = t - a * NE;
    float mn = 3.0e38f, mx = -3.0e38f;
    for (int s = 0; s < NS; ++s) {
        float v = out[NAS * NCOL + (size_t)(a * NS + s) * NCOL + a * NE + e];
        mn = fminf(mn, v);
        mx = fmaxf(mx, v);
    }
    float inv = 1.0f / (mx - mn + EPS);
    for (int s = 0; s < NS; ++s) {
        size_t off = (size_t)(a * NS + s) * NCOL + a * NE + e;
        float v = out[NAS * NCOL + off];
        simn[(a * NS + s) * NE + e] = (v - mn) * inv;
        bidx[(a * NS + s) * NE + e] = (int)out[off];
    }
}

// ---------------------------------------------------------------------------
// Kernel 6: vis_mat sum + nonzero count via Gram trick, block-reduced atomics
// ---------------------------------------------------------------------------
__global__ __launch_bounds__(256)
void k_vis(const float* __restrict__ simn, const int* __restrict__ bidx,
           const float* __restrict__ G, const int* __restrict__ ent,
           float* __restrict__ accum)
{
    __shared__ float rsum[256];
    __shared__ float rcnt[256];
    int t = blockIdx.x * 256 + threadIdx.x;
    // t -> (a, e, s, tt), total 16*12*32*32 = 196608
    int a  = t / (NE * NS * NS);
    int r  = t - a * (NE * NS * NS);
    int e  = r / (NS * NS);
    int r2 = r - e * (NS * NS);
    int s  = r2 / NS;
    int tt = r2 - s * NS;

    float val = 0.0f;
    if (e < ent[a] && s != tt) {
        float w1 = simn[(a * NS + s) * NE + e];
        float w2 = simn[(a * NS + tt) * NE + e];
        int   b1 = bidx[(a * NS + s) * NE + e];
        int   b2 = bidx[(a * NS + tt) * NE + e];
        val = 1.0f - w1 * w2 * G[b1 * NB + b2];
    }
    rsum[threadIdx.x] = val;
    rcnt[threadIdx.x] = (val != 0.0f) ? 1.0f : 0.0f;
    __syncthreads();
    for (int o = 128; o > 0; o >>= 1) {
        if (threadIdx.x < o) {
            rsum[threadIdx.x] += rsum[threadIdx.x + o];
            rcnt[threadIdx.x] += rcnt[threadIdx.x + o];
        }
        __syncthreads();
    }
    if (threadIdx.x == 0) {
        atomicAdd(&accum[0], rsum[0]);
        atomicAdd(&accum[1], rcnt[0]);
    }
}

// ---------------------------------------------------------------------------
// Kernel 7: per-(a, j) Smin/Smax over s of D_sim
// ---------------------------------------------------------------------------
__global__ void k_minmax(const float* __restrict__ out,
                         float* __restrict__ mn, float* __restrict__ mx)
{
    int t = blockIdx.x * 256 + threadIdx.x;
    if (t >= NA * NCOL) return;
    int a = t / NCOL, j = t - a * NCOL;
    float lo = 3.0e38f, hi = -3.0e38f;
    for (int s = 0; s < NS; ++s) {
        float v = out[NAS * NCOL + (size_t)(a * NS + s) * NCOL + j];
        lo = fminf(lo, v);
        hi = fmaxf(hi, v);
    }
    mn[t] = lo;
    mx[t] = hi;
}

// ---------------------------------------------------------------------------
// Kernel 8: Sf[a,s,a2] = sum_e S*((S-mn)/(mx-mn+EPS)) / div_vec[a2]
// ---------------------------------------------------------------------------
__global__ void k_sf(const float* __restrict__ out,
                     const float* __restrict__ mn, const float* __restrict__ mx,
                     const int* __restrict__ ent, float* __restrict__ Sf)
{
    int t = blockIdx.x * 256 + threadIdx.x;
    if (t >= NA * NS * NA) return;
    int a  = t / (NS * NA);
    int r  = t - a * (NS * NA);
    int s  = r / NA;
    int a2 = r - s * NA;
    float sum = 0.0f;
    for (int e = 0; e < NE; ++e) {
        int j = a2 * NE + e;
        float v  = out[NAS * NCOL + (size_t)(a * NS + s) * NCOL + j];
        float lo = mn[a * NCOL + j];
        float hi = mx[a * NCOL + j];
        sum += v * ((v - lo) / (hi - lo + EPS));
    }
    int dv = ent[a2];
    float div = (dv == 0) ? 1.0f : (float)dv;
    Sf[t] = sum / div;
}

// ---------------------------------------------------------------------------
// Kernel 9: margin terms + final scalar
// ---------------------------------------------------------------------------
__global__ void k_final(const float* __restrict__ Sf,
                        const float* __restrict__ accum,
                        float* __restrict__ out)
{
    __shared__ float r1[256];
    __shared__ float r2[256];
    float s1 = 0.0f, s2 = 0.0f;
    for (int idx = threadIdx.x; idx < NA * NS * NA; idx += 256) {
        int a  = idx / (NS * NA);
        int rr = idx - a * (NS * NA);
        int s  = rr / NA;
        int a2 = rr - s * NA;
        float v  = Sf[idx];
        float d1 = Sf[(a2 * NS + s) * NA + a2];   // Sf_diag[a2, s]
        float d2 = Sf[(a * NS + s) * NA + a];     // Sf_diag[a, s]
        s1 += fmaxf(v - d1 + DELTA, 0.0f);
        s2 += fmaxf(v - d2 + DELTA, 0.0f);
    }
    r1[threadIdx.x] = s1;
    r2[threadIdx.x] = s2;
    __syncthreads();
    for (int o = 128; o > 0; o >>= 1) {
        if (threadIdx.x < o) {
            r1[threadIdx.x] += r1[threadIdx.x + o];
            r2[threadIdx.x] += r2[threadIdx.x + o];
        }
        __syncthreads();
    }
    if (threadIdx.x == 0) {
        float frame_mean = (r1[0] + r2[0]) / (float)(NA * NS * NA);
        float cnt = accum[1];
        float vis_loss = (cnt != 0.0f) ? (accum[0] / cnt) : 0.0f;
        out[2 * NAS * NCOL] = (frame_mean + vis_loss) * 10.0f;
    }
}

__global__ void k_init(float* __restrict__ accum)
{
    accum[0] = 0.0f;
    accum[1] = 0.0f;
}

// ---------------------------------------------------------------------------
extern "C" void kernel_launch(void* const* d_in, const int* in_sizes, int n_in,
                              void* d_out, int out_size, void* d_ws, size_t ws_size,
                              hipStream_t stream)
{
    const float* vis  = (const float*)d_in[0];
    const float* word = (const float*)d_in[1];
    const int*   ent  = (const int*)d_in[2];
    float* out = (float*)d_out;

    // Workspace layout (floats)
    float* ws = (float*)d_ws;
    size_t o = 0;
    float* S    = ws + o; o += (size_t)NROW * NCOL;   // 9,830,400
    float* ninv = ws + o; o += 128;
    float* G    = ws + o; o += NB * NB + 48;          // 10,048
    float* simn = ws + o; o += NA * NS * NE;          // 6,144
    int*   bidx = (int*)(ws + o); o += NA * NS * NE;  // 6,144
    float* mn   = ws + o; o += NA * NCOL;             // 3,072
    float* mx   = ws + o; o += NA * NCOL;             // 3,072
    float* Sf   = ws + o; o += NA * NS * NA;          // 8,192
    float* accum = ws + o; o += 8;

    k_init<<<1, 1, 0, stream>>>(accum);
    k_gemm<<<NROW / 128, 256, 0, stream>>>(vis, word, ent, S);
    k_rowmax<<<(NAS * NCOL + 255) / 256, 256, 0, stream>>>(S, out);
    k_norms<<<NB, 256, 0, stream>>>(vis, ninv);
    k_gram<<<(NB * NB + 255) / 256, 256, 0, stream>>>(vis, ninv, G);
    k_simn<<<1, 192, 0, stream>>>(out, simn, bidx);
    k_vis<<<(NA * NE * NS * NS) / 256, 256, 0, stream>>>(simn, bidx, G, ent, accum);
    k_minmax<<<(NA * NCOL + 255) / 256, 256, 0, stream>>>(out, mn, mx);
    k_sf<<<(NA * NS * NA + 255) / 256, 256, 0, stream>>>(out, mn, mx, ent, Sf);
    k_final<<<1, 256, 0, stream>>>(Sf, accum, out);
}